// SGABlock_89395449299016
// MI455X (gfx1250) — compile-verified
//
#include <hip/hip_runtime.h>
#include <cstdint>

// Problem constants (from reference): B=1, C=32, D=48, H=96, W=192
#define C_ 32
#define D_ 48
#define H_ 96
#define W_ 192
#define HW_ (H_ * W_)          // 18432
#define DHW_ (D_ * H_ * W_)    // 884736
#define TOT_ (C_ * DHW_)       // 28311552
#define EPS_BN 1e-5f
#define NEG_INF (-3.402823466e38f)

#define XS 196   // padded row stride (floats) for W-rows: 16B-aligned, bank step 4
#define HS 100   // padded row stride (floats) for H-rows

// 16-byte int vector type, matching the async builtin's parameter type
typedef int v4i_ __attribute__((vector_size(4 * sizeof(int))));
typedef v4i_ __attribute__((address_space(1)))* gv4i_ptr;
typedef v4i_ __attribute__((address_space(3)))* lv4i_ptr;

// ---------- CDNA5 async global->LDS copy (tracked by ASYNCcnt) ----------
__device__ __forceinline__ void async_copy16(const float* gsrc, float* ldst) {
#if defined(__has_builtin) && __has_builtin(__builtin_amdgcn_global_load_async_to_lds_b128)
  __builtin_amdgcn_global_load_async_to_lds_b128(
      (gv4i_ptr)(uintptr_t)gsrc,
      (lv4i_ptr)(uint32_t)(uintptr_t)ldst,
      0, 0);
#else
  asm volatile("global_load_async_to_lds_b128 %0, %1, off"
               :: "v"((uint32_t)(uintptr_t)ldst), "v"(gsrc) : "memory");
#endif
}

__device__ __forceinline__ void async_wait0() {
#if defined(__has_builtin) && __has_builtin(__builtin_amdgcn_s_wait_asynccnt)
  __builtin_amdgcn_s_wait_asynccnt(0);
#else
  asm volatile("s_wait_asynccnt 0" ::: "memory");
#endif
}

// wave32 max-reduction (butterfly)
__device__ __forceinline__ float wave_max32(float m) {
  m = fmaxf(m, __shfl_xor(m, 16, 32));
  m = fmaxf(m, __shfl_xor(m, 8, 32));
  m = fmaxf(m, __shfl_xor(m, 4, 32));
  m = fmaxf(m, __shfl_xor(m, 2, 32));
  m = fmaxf(m, __shfl_xor(m, 1, 32));
  return m;
}

// =====================================================================
// Kernel A: W-direction scans (forward + reverse), one block per (c,h).
// Async-stages the x slab into LDS (overlapped with weight normalization),
// then wave0 scans forward and wave1 scans reverse concurrently.
// Writes max(a0,a1) into t (= d_out used as scratch).
// =====================================================================
__global__ __launch_bounds__(256) void sga_wscan_kernel(
    const float* __restrict__ x, const float* __restrict__ g, float* __restrict__ t)
{
  extern __shared__ float smem[];
  float* xs   = smem;                  // D_ * XS
  float* rs0  = xs  + D_ * XS;         // forward result slab
  float* rs1  = rs0 + D_ * XS;         // reverse result slab
  float* wbuf = rs1 + D_ * XS;         // 2 dirs * 5 * W_
  float* pbuf = wbuf + 2 * 5 * W_;     // 2 waves * 64

  const int c   = blockIdx.x / H_;
  const int h   = blockIdx.x % H_;
  const int tid = threadIdx.x;

  // ---- FIRST: kick off async staging of x slab x[c, :, h, :] into LDS ----
  // (fire-and-forget b128 DMA; proceeds while we normalize weights below)
  const float* xrow = x + (size_t)c * DHW_ + (size_t)h * W_;
  for (int s = tid; s < D_ * (W_ / 4); s += blockDim.x) {
    const int d  = s / (W_ / 4);
    const int w4 = s % (W_ / 4);
    async_copy16(xrow + (size_t)d * HW_ + w4 * 4, xs + d * XS + w4 * 4);
  }

  // ---- overlapped: load + L1-normalize weights for dirs 0 (fwd W), 1 (rev W) ----
  for (int s = tid; s < 2 * W_; s += blockDim.x) {
    const int q = s / W_;
    const int w = s % W_;
    const float* gp = g + ((size_t)(q * 160 + c * 5) * HW_) + (size_t)h * W_ + w;
    const float a0 = gp[0 * HW_], a1 = gp[1 * HW_], a2 = gp[2 * HW_],
                a3 = gp[3 * HW_], a4 = gp[4 * HW_];
    const float den = fmaxf(fabsf(a0) + fabsf(a1) + fabsf(a2) + fabsf(a3) + fabsf(a4), 1e-12f);
    const float inv = 1.0f / den;
    float* wb = wbuf + (q * 5) * W_ + w;
    wb[0 * W_] = a0 * inv; wb[1 * W_] = a1 * inv; wb[2 * W_] = a2 * inv;
    wb[3 * W_] = a3 * inv; wb[4 * W_] = a4 * inv;
  }

  async_wait0();          // own wave's ASYNCcnt drained
  __syncthreads();        // all waves' LDS writes visible

  const int wave = tid >> 5;
  const int lane = tid & 31;

  if (wave < 2) {                        // wave0: forward, wave1: reverse (concurrent)
    const int dir  = wave;
    float* rs      = dir ? rs1 : rs0;
    float* pb      = pbuf + wave * 64;
    const float* wq = wbuf + dir * 5 * W_;
    const int d0   = lane;
    const int d1   = 32 + lane;
    const bool has2 = (lane < 16);
    const int wstart = dir ? (W_ - 1) : 0;
    const int wstep  = dir ? -1 : 1;

    float p0 = xs[d0 * XS + wstart];
    float p1 = has2 ? xs[d1 * XS + wstart] : NEG_INF;
    rs[d0 * XS + wstart] = p0;
    if (has2) rs[d1 * XS + wstart] = p1;

    int w = wstart;
    for (int s = 1; s < W_; ++s) {
      w += wstep;
      const float w0w = wq[0 * W_ + w], w1w = wq[1 * W_ + w], w2w = wq[2 * W_ + w],
                  w3w = wq[3 * W_ + w], w4w = wq[4 * W_ + w];
      float m = has2 ? fmaxf(p0, p1) : p0;
      m = wave_max32(m);
      pb[d0] = p0;
      if (has2) pb[d1] = p1;
      __builtin_amdgcn_wave_barrier();           // wave-synchronous LDS exchange
      const float pm0 = d0 ? pb[d0 - 1] : 0.0f;
      const float pp0 = pb[d0 + 1];              // d0<=31 -> always valid
      const float xv0 = xs[d0 * XS + w];
      const float a = w0w * xv0 + w1w * p0 + w2w * pm0 + w3w * pp0 + w4w * m;
      rs[d0 * XS + w] = a;
      if (has2) {
        const float pm1 = pb[d1 - 1];
        const float pp1 = (d1 == D_ - 1) ? 0.0f : pb[d1 + 1];
        const float xv1 = xs[d1 * XS + w];
        const float b = w0w * xv1 + w1w * p1 + w2w * pm1 + w3w * pp1 + w4w * m;
        rs[d1 * XS + w] = b;
        p1 = b;
      }
      p0 = a;
    }
  }
  __syncthreads();

  // ---- write max(a0, a1) to t, coalesced rows ----
  float* trow = t + (size_t)c * DHW_ + (size_t)h * W_;
  for (int s = tid; s < D_ * W_; s += blockDim.x) {
    const int d = s / W_;
    const int w = s % W_;
    trow[(size_t)d * HW_ + w] = fmaxf(rs0[d * XS + w], rs1[d * XS + w]);
  }
}

// =====================================================================
// Kernel B: H-direction scans, one block per (c, pair-of-w). 4 waves =
// 4 (w,dir) scans, all lanes busy. Merges max into t, emits BN partials.
// x columns are read directly (x fits in the 192MB L2), with prefetch.
// =====================================================================
__global__ __launch_bounds__(128) void sga_hscan_kernel(
    const float* __restrict__ x, const float* __restrict__ g,
    float* __restrict__ t, float* __restrict__ partials)
{
  extern __shared__ float smem[];
  float* rslab = smem;                     // 4 * D_*HS
  float* wbuf  = rslab + 4 * D_ * HS;      // 2 w * 2 dir * 5 * H_
  float* pbuf  = wbuf + 2 * 2 * 5 * H_;    // 4 waves * 64
  float* red1  = pbuf + 4 * 64;            // 128
  float* red2  = red1 + 128;               // 128

  const int c   = blockIdx.x / (W_ / 2);
  const int wp  = blockIdx.x % (W_ / 2);
  const int w0  = wp * 2;
  const int tid = threadIdx.x;

  // ---- load + L1-normalize weights for dirs 2 (fwd H) and 3 (rev H) ----
  for (int s = tid; s < 2 * 2 * H_; s += blockDim.x) {
    const int j = s / (2 * H_);
    const int q = (s / H_) & 1;
    const int h = s % H_;
    const float* gp = g + ((size_t)((2 + q) * 160 + c * 5) * HW_) + (size_t)h * W_ + (w0 + j);
    const float a0 = gp[0 * HW_], a1 = gp[1 * HW_], a2 = gp[2 * HW_],
                a3 = gp[3 * HW_], a4 = gp[4 * HW_];
    const float den = fmaxf(fabsf(a0) + fabsf(a1) + fabsf(a2) + fabsf(a3) + fabsf(a4), 1e-12f);
    const float inv = 1.0f / den;
    float* wb = wbuf + ((j * 2 + q) * 5) * H_ + h;
    wb[0 * H_] = a0 * inv; wb[1 * H_] = a1 * inv; wb[2 * H_] = a2 * inv;
    wb[3 * H_] = a3 * inv; wb[4 * H_] = a4 * inv;
  }
  __syncthreads();

  const int wave = tid >> 5;        // 0..3: (j = wave>>1, dir = wave&1)
  const int lane = tid & 31;
  const int j    = wave >> 1;
  const int dir  = wave & 1;
  const int d0   = lane;
  const int d1   = 32 + lane;
  const bool has2 = (lane < 16);
  const float* wq = wbuf + ((j * 2 + dir) * 5) * H_;
  float* rs = rslab + wave * (D_ * HS);
  float* pb = pbuf + wave * 64;
  const float* xb = x + (size_t)c * DHW_ + (w0 + j);

  const int hstart = dir ? (H_ - 1) : 0;
  const int hstep  = dir ? -1 : 1;
  float p0 = xb[(size_t)d0 * HW_ + (size_t)hstart * W_];
  float p1 = has2 ? xb[(size_t)d1 * HW_ + (size_t)hstart * W_] : NEG_INF;
  rs[d0 * HS + hstart] = p0;
  if (has2) rs[d1 * HS + hstart] = p1;

  int h = hstart;
  for (int s = 1; s < H_; ++s) {
    h += hstep;
    const float xv0 = xb[(size_t)d0 * HW_ + (size_t)h * W_];
    const float xv1 = has2 ? xb[(size_t)d1 * HW_ + (size_t)h * W_] : 0.0f;
    if (s + 1 < H_)
      __builtin_prefetch(xb + (size_t)d0 * HW_ + (size_t)(h + hstep) * W_, 0, 1);
    const float w0w = wq[0 * H_ + h], w1w = wq[1 * H_ + h], w2w = wq[2 * H_ + h],
                w3w = wq[3 * H_ + h], w4w = wq[4 * H_ + h];
    float m = has2 ? fmaxf(p0, p1) : p0;
    m = wave_max32(m);
    pb[d0] = p0;
    if (has2) pb[d1] = p1;
    __builtin_amdgcn_wave_barrier();
    const float pm0 = d0 ? pb[d0 - 1] : 0.0f;
    const float pp0 = pb[d0 + 1];
    const float a = w0w * xv0 + w1w * p0 + w2w * pm0 + w3w * pp0 + w4w * m;
    rs[d0 * HS + h] = a;
    if (has2) {
      const float pm1 = pb[d1 - 1];
      const float pp1 = (d1 == D_ - 1) ? 0.0f : pb[d1 + 1];
      const float b = w0w * xv1 + w1w * p1 + w2w * pm1 + w3w * pp1 + w4w * m;
      rs[d1 * HS + h] = b;
      p1 = b;
    }
    p0 = a;
  }
  __syncthreads();

  // ---- merge 4-direction max into t, accumulate per-block BN partials ----
  const float* rsF0 = rslab + 0 * (D_ * HS);
  const float* rsR0 = rslab + 1 * (D_ * HS);
  const float* rsF1 = rslab + 2 * (D_ * HS);
  const float* rsR1 = rslab + 3 * (D_ * HS);
  float s1 = 0.0f, s2 = 0.0f;
  for (int s = tid; s < D_ * H_; s += blockDim.x) {
    const int d  = s / H_;
    const int hh = s % H_;
    const size_t base = (((size_t)c * D_ + d) * H_ + hh) * W_ + w0;
    float2 tv = *reinterpret_cast<float2*>(t + base);
    const int o = d * HS + hh;
    const float v0 = fmaxf(tv.x, fmaxf(rsF0[o], rsR0[o]));
    const float v1 = fmaxf(tv.y, fmaxf(rsF1[o], rsR1[o]));
    tv.x = v0; tv.y = v1;
    *reinterpret_cast<float2*>(t + base) = tv;
    s1 += v0 + v1;
    s2 += v0 * v0 + v1 * v1;
  }
  red1[tid] = s1; red2[tid] = s2;
  __syncthreads();
  for (int off = 64; off > 0; off >>= 1) {
    if (tid < off) { red1[tid] += red1[tid + off]; red2[tid] += red2[tid + off]; }
    __syncthreads();
  }
  if (tid == 0) {
    partials[2 * blockIdx.x]     = red1[0];
    partials[2 * blockIdx.x + 1] = red2[0];
  }
}

// =====================================================================
// Kernel C: fold partials -> per-channel scale/bias (deterministic order)
// =====================================================================
__global__ void sga_stats_kernel(const float* __restrict__ partials,
                                 const float* __restrict__ gamma,
                                 const float* __restrict__ beta,
                                 float* __restrict__ params)
{
  const int c = threadIdx.x;
  if (c < C_) {
    double s1 = 0.0, s2 = 0.0;
    for (int wp = 0; wp < (W_ / 2); ++wp) {
      s1 += (double)partials[2 * (c * (W_ / 2) + wp)];
      s2 += (double)partials[2 * (c * (W_ / 2) + wp) + 1];
    }
    const double n    = (double)DHW_;
    const float mean  = (float)(s1 / n);
    const float var   = (float)(s2 / n) - mean * mean;
    const float inv   = 1.0f / sqrtf(var + EPS_BN);
    const float sc    = gamma[c] * inv;
    params[c]      = sc;
    params[C_ + c] = beta[c] - mean * sc;
  }
}

// =====================================================================
// Kernel D: out = relu(scale[c]*t + bias[c] + x), in place, float4
// =====================================================================
__global__ __launch_bounds__(256) void sga_finalize_kernel(
    const float* __restrict__ x, const float* __restrict__ params, float* __restrict__ t)
{
  const size_t i4 = (size_t)blockIdx.x * blockDim.x + threadIdx.x;
  if (i4 >= (size_t)(TOT_ / 4)) return;
  const int c = (int)(i4 / (DHW_ / 4));
  const float sc = params[c];
  const float bi = params[C_ + c];
  float4 tv = reinterpret_cast<const float4*>(t)[i4];
  const float4 xv = reinterpret_cast<const float4*>(x)[i4];
  tv.x = fmaxf(0.0f, fmaf(sc, tv.x, bi) + xv.x);
  tv.y = fmaxf(0.0f, fmaf(sc, tv.y, bi) + xv.y);
  tv.z = fmaxf(0.0f, fmaf(sc, tv.z, bi) + xv.z);
  tv.w = fmaxf(0.0f, fmaf(sc, tv.w, bi) + xv.w);
  reinterpret_cast<float4*>(t)[i4] = tv;
}

// =====================================================================
extern "C" void kernel_launch(void* const* d_in, const int* in_sizes, int n_in,
                              void* d_out, int out_size, void* d_ws, size_t ws_size,
                              hipStream_t stream) {
  (void)in_sizes; (void)n_in; (void)out_size; (void)ws_size;
  const float* x     = (const float*)d_in[0];
  const float* g     = (const float*)d_in[1];
  const float* gamma = (const float*)d_in[2];
  const float* beta  = (const float*)d_in[3];
  float* out = (float*)d_out;                  // doubles as the 113MB scan scratch
  float* wsf = (float*)d_ws;
  float* partials = wsf;                       // 2 * 3072 floats
  float* params   = wsf + 8192;                // 2 * 32 floats

  const size_t smemA = (size_t)(3 * D_ * XS + 2 * 5 * W_ + 2 * 64) * sizeof(float);
  sga_wscan_kernel<<<C_ * H_, 256, smemA, stream>>>(x, g, out);

  const size_t smemB = (size_t)(4 * D_ * HS + 2 * 2 * 5 * H_ + 4 * 64 + 2 * 128) * sizeof(float);
  sga_hscan_kernel<<<C_ * (W_ / 2), 128, smemB, stream>>>(x, g, out, partials);

  sga_stats_kernel<<<1, 64, 0, stream>>>(partials, gamma, beta, params);

  sga_finalize_kernel<<<(TOT_ / 4) / 256, 256, 0, stream>>>(x, params, out);
}